// soft_attention_90151363543173
// MI455X (gfx1250) — compile-verified
//
#include <hip/hip_runtime.h>
#include <math.h>

// ---------------- problem geometry ----------------
constexpr int D        = 512;    // feature dim
constexpr int L        = 4096;   // sequence length
constexpr int NBATCH   = 128;    // batch
constexpr int LSPLIT   = 8;      // l-chunks per batch row (grid.x)
constexpr int WAVES    = 8;      // waves per block
constexpr int THREADS  = WAVES * 32;
constexpr int ROWS_PER_BLOCK = L / LSPLIT;             // 512
constexpr int ROWS_PER_WAVE  = ROWS_PER_BLOCK / WAVES; // 64
constexpr int BUF_ROWS = 4;                            // rows per async buffer
constexpr int NFILL    = ROWS_PER_WAVE / BUF_ROWS;     // 16 fills per wave
constexpr int FILL_OPS = BUF_ROWS * D * 4 / 512;       // async b128 ops per fill
static_assert(FILL_OPS == 16, "s_wait_asynccnt literal below assumes 16 ops/fill");

typedef __attribute__((ext_vector_type(2))) float v2f;
typedef __attribute__((ext_vector_type(8))) float v8f;

// ---------------- helpers ----------------
typedef __attribute__((address_space(3))) const void* lds_cptr;

__device__ __forceinline__ unsigned ldsOffset(const void* p) {
  // generic -> LDS addrspace -> 32-bit LDS byte offset
  return (unsigned)(unsigned long long)(lds_cptr)p;
}

__device__ __forceinline__ float ftanh(float x) {
#if __has_builtin(__builtin_amdgcn_tanhf)
  return __builtin_amdgcn_tanhf(x);   // v_tanh_f32 on gfx1250
#else
  return tanhf(x);
#endif
}

// Async-copy BUF_ROWS*D floats (8 KB) from global to LDS.
// Each op: 32 lanes x 16 B = 512 B contiguous; 16 ops cover the buffer.
// Non-temporal hint: x is streamed exactly once, keep it out of L2.
__device__ __forceinline__ void asyncFillBuf(const float* gBase, unsigned ldsBase, int lane) {
  unsigned long long g = (unsigned long long)gBase + (unsigned)(lane * 16);
  unsigned           l = ldsBase + (unsigned)(lane * 16);
#pragma unroll
  for (int u = 0; u < FILL_OPS; ++u) {
    asm volatile("global_load_async_to_lds_b128 %0, %1, off th:TH_LOAD_NT"
                 :: "v"(l + (unsigned)(u * 512)),
                    "v"(g + (unsigned long long)(u * 512))
                 : "memory");
  }
}

// ---------------- kernel 1: fused tanh-score + online softmax partials ----------------
__global__ __launch_bounds__(THREADS)
void attn_partial_kernel(const float* __restrict__ x,
                         const float* __restrict__ linear,
                         float* __restrict__ accW,   // [NBATCH*LSPLIT][D]
                         float* __restrict__ mW,     // [NBATCH*LSPLIT]
                         float* __restrict__ sW) {   // [NBATCH*LSPLIT]
  __shared__ __align__(16) float tile[WAVES * 2 * BUF_ROWS * D]; // 128 KB double buffers
  __shared__ __align__(16) float cAcc[WAVES * D];                // 16 KB wave-combine
  __shared__ float cM[WAVES], cS[WAVES];

  const int tid  = threadIdx.x;
  const int w    = tid >> 5;
  const int lane = tid & 31;
  const int c    = blockIdx.x;   // l-chunk
  const int n    = blockIdx.y;   // batch row

  const float* xw = x + ((size_t)n * L + (size_t)c * ROWS_PER_BLOCK
                         + (size_t)w * ROWS_PER_WAVE) * D;
  float* myTile = tile + w * (2 * BUF_ROWS * D);
  const unsigned off0 = ldsOffset(myTile);
  const unsigned off1 = off0 + (unsigned)(BUF_ROWS * D * sizeof(float));

  // per-lane slice of `linear`: d = k*128 + lane*4 + {0..3}
  float4 lin4[4];
#pragma unroll
  for (int k = 0; k < 4; ++k) lin4[k] = ((const float4*)linear)[k * 32 + lane];

  float4 acc4[4];
#pragma unroll
  for (int k = 0; k < 4; ++k) acc4[k] = make_float4(0.f, 0.f, 0.f, 0.f);
  float m = -__builtin_inff();
  float s = 0.f;

  asyncFillBuf(xw, off0, lane);                       // prime buffer 0

  for (int t = 0; t < NFILL; ++t) {
    const float* buf = myTile + (t & 1) * (BUF_ROWS * D);
    if (t + 1 < NFILL) {
      // start filling the other buffer, then wait only for the current one
      asyncFillBuf(xw + (size_t)(t + 1) * BUF_ROWS * D, (t & 1) ? off0 : off1, lane);
      asm volatile("s_wait_asynccnt 16" ::: "memory");
    } else {
      asm volatile("s_wait_asynccnt 0" ::: "memory");
    }

#pragma unroll
    for (int r = 0; r < BUF_ROWS; ++r) {
      const float4* rowp = (const float4*)(buf + r * D);
      float4 xv[4];
#pragma unroll
      for (int k = 0; k < 4; ++k) xv[k] = rowp[k * 32 + lane];  // ds_load_b128

      // score_l = sum_d tanh(x[l,d]) * linear[d]
      float dot = 0.f;
#pragma unroll
      for (int k = 0; k < 4; ++k) {
        dot += ftanh(xv[k].x) * lin4[k].x;
        dot += ftanh(xv[k].y) * lin4[k].y;
        dot += ftanh(xv[k].z) * lin4[k].z;
        dot += ftanh(xv[k].w) * lin4[k].w;
      }
#pragma unroll
      for (int o = 16; o >= 1; o >>= 1) dot += __shfl_xor(dot, o, 32);

      // branchless online-softmax update (raw x accumulated)
      const float nm = fmaxf(m, dot);
      const float cr = __expf(m - nm);    // first iter: exp(-inf)=0
      const float p  = __expf(dot - nm);
      s = s * cr + p;
      m = nm;
#pragma unroll
      for (int k = 0; k < 4; ++k) {
        acc4[k].x = acc4[k].x * cr + p * xv[k].x;
        acc4[k].y = acc4[k].y * cr + p * xv[k].y;
        acc4[k].z = acc4[k].z * cr + p * xv[k].z;
        acc4[k].w = acc4[k].w * cr + p * xv[k].w;
      }
    }
  }

  // ---- publish per-wave partial states ----
#pragma unroll
  for (int k = 0; k < 4; ++k) ((float4*)(cAcc + w * D))[k * 32 + lane] = acc4[k];
  if (lane == 0) { cM[w] = m; cS[w] = s; }
  __syncthreads();

  float M = -__builtin_inff();
#pragma unroll
  for (int w2 = 0; w2 < WAVES; ++w2) M = fmaxf(M, cM[w2]);
  float f[WAVES];
  float stot = 0.f;
#pragma unroll
  for (int w2 = 0; w2 < WAVES; ++w2) { f[w2] = __expf(cM[w2] - M); stot += f[w2] * cS[w2]; }

  const size_t part = (size_t)n * LSPLIT + c;

  // ---- rank-8 combine acc_b[d] = sum_w f_w * cAcc[w][d] via V_WMMA_F32_16X16X4_F32 ----
  // A: all 16 M-rows identical = f-weights; B: the 8 wave accumulators.
  // A and B use the same (lane-half, VGPR)->K slot mapping, so the contraction
  // is invariant to the hardware's exact K labeling. Full f32 precision.
  const int half = lane >> 4;          // slot map: (lo,v0)=K0 (lo,v1)=K1 (hi,v0)=K2 (hi,v1)=K3
  const int nl   = lane & 15;
  v2f a_lo, a_hi;
  a_lo.x = f[half * 2 + 0];  a_lo.y = f[half * 2 + 1];       // K = 0..3
  a_hi.x = f[4 + half * 2];  a_hi.y = f[4 + half * 2 + 1];   // K = 4..7
#pragma unroll
  for (int t16 = 0; t16 < 4; ++t16) {
    const int d0 = w * 64 + t16 * 16;   // each wave owns 64 d-columns
    v2f b_lo, b_hi;
    b_lo.x = cAcc[(half * 2 + 0) * D + d0 + nl];
    b_lo.y = cAcc[(half * 2 + 1) * D + d0 + nl];
    b_hi.x = cAcc[(4 + half * 2 + 0) * D + d0 + nl];
    b_hi.y = cAcc[(4 + half * 2 + 1) * D + d0 + nl];
    v8f cmat = {};
    cmat = __builtin_amdgcn_wmma_f32_16x16x4_f32(false, a_lo, false, b_lo,
                                                 (short)0, cmat, false, false);
    cmat = __builtin_amdgcn_wmma_f32_16x16x4_f32(false, a_hi, false, b_hi,
                                                 (short)0, cmat, false, false);
    // every D row is identical; row M=0 lives in VGPR0 of lanes 0..15 (N = lane)
    if (lane < 16) accW[part * D + d0 + lane] = cmat[0];
  }
  if (tid == 0) { mW[part] = M; sW[part] = stot; }
}

// ---------------- kernel 2: merge l-chunks, normalize, tanh ----------------
__global__ __launch_bounds__(256)
void attn_combine_kernel(const float* __restrict__ accW,
                         const float* __restrict__ mW,
                         const float* __restrict__ sW,
                         float* __restrict__ out) {
  const int n   = blockIdx.x;
  const int tid = threadIdx.x;

  float M = -__builtin_inff();
#pragma unroll
  for (int c = 0; c < LSPLIT; ++c) M = fmaxf(M, mW[n * LSPLIT + c]);
  float f[LSPLIT];
  float stot = 0.f;
#pragma unroll
  for (int c = 0; c < LSPLIT; ++c) {
    f[c] = __expf(mW[n * LSPLIT + c] - M);
    stot += f[c] * sW[n * LSPLIT + c];
  }
  const float inv = 1.f / stot;

  const int d0 = tid * 2;
  float2 a = make_float2(0.f, 0.f);
#pragma unroll
  for (int c = 0; c < LSPLIT; ++c) {
    const float* ap = accW + ((size_t)n * LSPLIT + c) * D;
    a.x += f[c] * ap[d0];
    a.y += f[c] * ap[d0 + 1];
  }
  ((float2*)(out + (size_t)n * D))[tid] =
      make_float2(ftanh(a.x * inv), ftanh(a.y * inv));
}

// ---------------- host entry ----------------
extern "C" void kernel_launch(void* const* d_in, const int* in_sizes, int n_in,
                              void* d_out, int out_size, void* d_ws, size_t ws_size,
                              hipStream_t stream) {
  const float* x      = (const float*)d_in[0];   // (128, 4096, 512) f32
  const float* linear = (const float*)d_in[1];   // (1, 512) f32
  float* out = (float*)d_out;                    // (128, 512) f32

  float* accW = (float*)d_ws;                                  // 2 MB
  float* mW   = accW + (size_t)NBATCH * LSPLIT * D;            // 4 KB
  float* sW   = mW + (size_t)NBATCH * LSPLIT;                  // 4 KB

  dim3 grid(LSPLIT, NBATCH);
  attn_partial_kernel<<<grid, THREADS, 0, stream>>>(x, linear, accW, mW, sW);
  attn_combine_kernel<<<NBATCH, 256, 0, stream>>>(accW, mW, sW, out);
}